// generator_loss_function_54331336294508
// MI455X (gfx1250) — compile-verified
//
#include <hip/hip_runtime.h>
#include <hip/hip_bf16.h>

typedef __attribute__((ext_vector_type(2))) float v2f;
typedef __attribute__((ext_vector_type(8))) float v8f;

#define IMG 512
#define OUT_DIM 506          // 512 - (7-1)
#define LDS_W 72             // input region: 38 rows x 72 cols (incl. 2 pad cols for K-pad reads)
#define LDS_H 38

__device__ __forceinline__ v8f wmma4(v2f a, v2f b, v8f c) {
  // D = A(16x4, f32) * B(4x16, f32) + C(16x16, f32)
  return __builtin_amdgcn_wmma_f32_16x16x4_f32(false, a, false, b, (short)0, c, false, false);
}

__global__ void init_ws(double* __restrict__ acc) {
  int i = blockIdx.x * blockDim.x + threadIdx.x;
  if (i < 768) acc[i] = 0.0;
}

// One block: 32x64 output tile of one frame's SSIM map (8 waves, each a 16x16 tile).
// Fused: L_rec partial sums over the exclusively-owned 32x64 input sub-tile.
__global__ __launch_bounds__(256)
void ssim_rec_kernel(const float* __restrict__ gen, const float* __restrict__ tgt,
                     double* __restrict__ acc) {
  __shared__ float sx[LDS_H * LDS_W];    // 255 * gen tile
  __shared__ float sy[LDS_H * LDS_W];    // 255 * tgt tile
  __shared__ float hs[8 * 24 * 16];      // per-wave H scratch (22 rows + 2 zero pad rows)
  __shared__ float red[3][8];

  const int tid  = threadIdx.x;
  const int lane = tid & 31;
  const int w    = tid >> 5;
  const int l15  = lane & 15;
  const int half = lane >> 4;            // lanes 16..31 hold K+2 per WMMA f32 layout

  const int row0 = blockIdx.y * 32;
  const int col0 = blockIdx.x * 64;
  const size_t fbase = (size_t)blockIdx.z * (size_t)(IMG * IMG);

  // Zero the two pad rows (22,23) of every wave's H scratch: exactly 256 elements.
  {
    int rem = tid & 31;
    hs[(tid >> 5) * 384 + (22 + (rem >> 4)) * 16 + (rem & 15)] = 0.0f;
  }

  // Cooperative load of 38x72 region (zero-filled outside image), fused L_rec.
  float recAbs = 0.0f, recSq = 0.0f;
  for (int idx = tid; idx < LDS_H * LDS_W; idx += 256) {
    int r = idx / LDS_W, c = idx - r * LDS_W;
    int gr = row0 + r, gc = col0 + c;
    float xv = 0.0f, yv = 0.0f;
    if (gr < IMG && gc < IMG) {
      size_t o = fbase + (size_t)gr * IMG + gc;
      xv = gen[o];
      yv = tgt[o];
    }
    if (r < 32 && c < 64) {              // owned sub-tile: always inside the image
      float d = xv - yv;
      recAbs += fabsf(d);
      recSq  += d * d;
    }
    sx[idx] = 255.0f * xv;
    sy[idx] = 255.0f * yv;
  }
  __syncthreads();

  // This wave's tile.
  const int wr = w >> 2, wc = w & 3;     // 2x4 waves
  const int lr0 = wr * 16, lc0 = wc * 16;
  const int orow = row0 + lr0, ocol = col0 + lc0;

  // Banded 7-ones matrix element per lane per K-chunk; identical for Sh (step1 B)
  // and Sv (step2 A) under the WMMA f32 A/B lane layouts.
  v2f band[6];
#pragma unroll
  for (int kc = 0; kc < 6; ++kc) {
    int k = kc * 4 + 2 * half;
    band[kc].x = (k     >= l15 && k     < l15 + 7) ? 1.0f : 0.0f;
    band[kc].y = (k + 1 >= l15 && k + 1 < l15 + 7) ? 1.0f : 0.0f;
  }

  // ---- Step 1: horizontal 7-sum. H(22x16) = In(22x24) * Sh(24x16), two 16-row M tiles.
  v8f H[10] = {};                        // [2*q + mtile], q in {x, y, xx, yy, xy}
  const int m1c = (16 + l15) < 22 ? (16 + l15) : 21;   // clamp pad rows (discarded later)
  const int rA0 = lr0 + l15;
  const int rA1 = lr0 + m1c;
#pragma unroll
  for (int kc = 0; kc < 6; ++kc) {
    int ka = kc * 4 + 2 * half;
    int off0 = rA0 * LDS_W + lc0 + ka;
    int off1 = rA1 * LDS_W + lc0 + ka;
    float x00 = sx[off0], x01 = sx[off0 + 1];
    float y00 = sy[off0], y01 = sy[off0 + 1];
    float x10 = sx[off1], x11 = sx[off1 + 1];
    float y10 = sy[off1], y11 = sy[off1 + 1];
    v2f b = band[kc];
    v2f a;
    a.x = x00;       a.y = x01;       H[0] = wmma4(a, b, H[0]);
    a.x = x10;       a.y = x11;       H[1] = wmma4(a, b, H[1]);
    a.x = y00;       a.y = y01;       H[2] = wmma4(a, b, H[2]);
    a.x = y10;       a.y = y11;       H[3] = wmma4(a, b, H[3]);
    a.x = x00 * x00; a.y = x01 * x01; H[4] = wmma4(a, b, H[4]);
    a.x = x10 * x10; a.y = x11 * x11; H[5] = wmma4(a, b, H[5]);
    a.x = y00 * y00; a.y = y01 * y01; H[6] = wmma4(a, b, H[6]);
    a.x = y10 * y10; a.y = y11 * y11; H[7] = wmma4(a, b, H[7]);
    a.x = x00 * y00; a.y = x01 * y01; H[8] = wmma4(a, b, H[8]);
    a.x = x10 * y10; a.y = x11 * y11; H[9] = wmma4(a, b, H[9]);
  }

  // ---- Step 2: vertical 7-sum per quantity. Out(16x16) = Sv(16x24) * H(24x16).
  const int wb = w * 384;
  v8f O[5];
#pragma unroll
  for (int q = 0; q < 5; ++q) {
#pragma unroll
    for (int i = 0; i < 8; ++i) {
      int rr = (lane < 16) ? i : (8 + i);          // C/D layout row
      hs[wb + rr * 16 + l15] = H[2 * q][i];        // H rows 0..15
      int rr1 = 16 + rr;
      if (rr1 < 22) hs[wb + rr1 * 16 + l15] = H[2 * q + 1][i];  // H rows 16..21
    }
    v8f o = {};
#pragma unroll
    for (int kc = 0; kc < 6; ++kc) {
      int kb = kc * 4 + 2 * half;
      v2f bb;
      bb.x = hs[wb + kb * 16 + l15];
      bb.y = hs[wb + (kb + 1) * 16 + l15];
      o = wmma4(band[kc], bb, o);                  // Sv band * H
    }
    O[q] = o;
  }

  // ---- SSIM map from 7x7 window sums (sums, not means: fold 1/49 + N/(N-1) in).
  const float inv49 = 1.0f / 49.0f;
  const float covn  = 49.0f / 48.0f;
  const float C1 = 6.5025f;     // (0.01*255)^2
  const float C2 = 58.5225f;    // (0.03*255)^2
  float ssimSum = 0.0f;
#pragma unroll
  for (int i = 0; i < 8; ++i) {
    float ux  = O[0][i] * inv49, uy  = O[1][i] * inv49;
    float uxx = O[2][i] * inv49, uyy = O[3][i] * inv49, uxy = O[4][i] * inv49;
    float vx  = covn * (uxx - ux * ux);
    float vy  = covn * (uyy - uy * uy);
    float vxy = covn * (uxy - ux * uy);
    float A1 = 2.0f * ux * uy + C1;
    float A2 = 2.0f * vxy + C2;
    float B1 = ux * ux + uy * uy + C1;
    float B2 = vx + vy + C2;
    float S = (A1 * A2) / (B1 * B2);
    int gr = orow + ((lane < 16) ? i : (8 + i));
    int gc = ocol + l15;
    if (gr < OUT_DIM && gc < OUT_DIM) ssimSum += S;
  }

  // ---- wave32 + block reduction, one f64 atomic per block into 256 striped slots.
  float v0 = ssimSum, v1 = recAbs, v2 = recSq;
#pragma unroll
  for (int off = 16; off > 0; off >>= 1) {
    v0 += __shfl_xor(v0, off, 32);
    v1 += __shfl_xor(v1, off, 32);
    v2 += __shfl_xor(v2, off, 32);
  }
  if (lane == 0) { red[0][w] = v0; red[1][w] = v1; red[2][w] = v2; }
  __syncthreads();
  if (tid == 0) {
    float s0 = 0.0f, s1 = 0.0f, s2 = 0.0f;
    for (int i = 0; i < 8; ++i) { s0 += red[0][i]; s1 += red[1][i]; s2 += red[2][i]; }
    int slot = (blockIdx.x + 8 * blockIdx.y + 128 * blockIdx.z) & 255;
    atomicAdd(&acc[slot],       (double)s0);  // ssim
    atomicAdd(&acc[256 + slot], (double)s1);  // sum |d|
    atomicAdd(&acc[512 + slot], (double)s2);  // sum d^2
  }
}

__global__ void finalize_kernel(const double* __restrict__ acc,
                                const float* __restrict__ genD, int nD,
                                float* __restrict__ out,
                                double nTotal, double nSsim) {
  if (threadIdx.x == 0 && blockIdx.x == 0) {
    double ss = 0.0, sa = 0.0, sq = 0.0;
    for (int i = 0; i < 256; ++i) { ss += acc[i]; sa += acc[256 + i]; sq += acc[512 + i]; }
    double Lrec  = sa / nTotal + sq / nTotal;
    double Lssim = ss / nSsim;
    double gd = 0.0;
    for (int i = 0; i < nD; ++i) gd += (double)genD[i];
    double Ladv = -(gd / (double)nD);
    double Ltot = Lrec + 0.01 * (1.0 - Lssim) + 0.0001 * Ladv;
    out[0] = (float)Ltot;
    out[1] = (float)Lrec;
    out[2] = (float)Lssim;
    out[3] = (float)Ladv;
  }
}

extern "C" void kernel_launch(void* const* d_in, const int* in_sizes, int n_in,
                              void* d_out, int out_size, void* d_ws, size_t ws_size,
                              hipStream_t stream) {
  (void)n_in; (void)out_size; (void)ws_size;
  const float* gen = (const float*)d_in[0];
  const float* tgt = (const float*)d_in[1];
  const float* gD  = (const float*)d_in[2];
  float* out  = (float*)d_out;
  double* acc = (double*)d_ws;      // 768 doubles: [ssim|abs|sq] x 256 striped slots

  const int frames = in_sizes[0] / (IMG * IMG);   // 160 for the reference shapes
  const int nD = in_sizes[2];

  hipLaunchKernelGGL(init_ws, dim3(3), dim3(256), 0, stream, acc);

  dim3 grid(8, 16, frames);          // 8 x 64-col tiles, 16 x 32-row tiles per frame
  hipLaunchKernelGGL(ssim_rec_kernel, grid, dim3(256), 0, stream, gen, tgt, acc);

  const double nTotal = (double)in_sizes[0];
  const double nSsim  = (double)frames * (double)OUT_DIM * (double)OUT_DIM;
  hipLaunchKernelGGL(finalize_kernel, dim3(1), dim3(32), 0, stream,
                     acc, gD, nD, out, nTotal, nSsim);
}